// GATLayer_2345052144365
// MI455X (gfx1250) — compile-verified
//
#include <hip/hip_runtime.h>
#include <hip/hip_bf16.h>

#define B 8
#define N 1024
#define IN_F 128
#define H 4
#define FH 32
#define OUTF (H * FH)          // 128
#define NEG_SLOPE 0.2f
#define NCHUNK 128
#define APAD 130               // padded j-stride for LDS alpha tile (bank-conflict free)

typedef __attribute__((ext_vector_type(2))) float v2f;
typedef __attribute__((ext_vector_type(8))) float v8f;

// ---------------------------------------------------------------------------
// Kernel 1: h[b,n,o] = sum_i x[b,n,i] * W[o,i]   (rows = b*N+n flattened)
// One block = 16 rows, 8 waves each produce a 16x16 column tile via f32 WMMA.
// ---------------------------------------------------------------------------
__global__ __launch_bounds__(256) void k_h(const float* __restrict__ x,
                                           const float* __restrict__ W,
                                           float* __restrict__ hbuf) {
    __shared__ float xs[16][APAD];   // 16 x 128 x-tile, padded
    const int tid  = threadIdx.x;
    const int row0 = blockIdx.x * 16;

    for (int idx = tid; idx < 16 * IN_F; idx += 256) {
        int rr = idx >> 7, cc = idx & 127;
        xs[rr][cc] = x[(size_t)(row0 + rr) * IN_F + cc];
    }
    __syncthreads();

    const int wid = tid >> 5;        // 0..7 -> column tile
    const int L   = tid & 31;
    const int o0  = wid * 16;
    const int mr  = L & 15;          // M row / N col within tile
    const int kb  = (L >> 4) * 2;    // K offset for f32 A/B operands
    const int hi  = L >> 4;

    v8f acc = {};
    for (int k0 = 0; k0 < IN_F; k0 += 4) {
        v2f a, bb;
        a.x  = xs[mr][k0 + kb];
        a.y  = xs[mr][k0 + kb + 1];
        // B[k][o] = W[o0+n][k], n = L&15
        bb.x = W[(size_t)(o0 + mr) * IN_F + k0 + kb];
        bb.y = W[(size_t)(o0 + mr) * IN_F + k0 + kb + 1];
        acc = __builtin_amdgcn_wmma_f32_16x16x4_f32(false, a, false, bb,
                                                    (short)0, acc, false, false);
    }

    // D layout: VGPR r, lanes 0-15 -> M=r, lanes 16-31 -> M=r+8, col = L&15
    for (int r = 0; r < 8; ++r) {
        int m = r + 8 * hi;
        hbuf[(size_t)(row0 + m) * OUTF + o0 + (L & 15)] = acc[r];
    }
}

// ---------------------------------------------------------------------------
// Kernel 2: src/dst per (node, head): 32-length dots against attn rows.
// attn layout: (H, 2*FH+1); a_src = [0:32), a_dst = [32:64), a_e = [64].
// ---------------------------------------------------------------------------
__global__ __launch_bounds__(256) void k_sd(const float* __restrict__ hbuf,
                                            const float* __restrict__ attn,
                                            float* __restrict__ srcb,
                                            float* __restrict__ dstb) {
    int idx = blockIdx.x * 256 + threadIdx.x;
    if (idx >= B * N * H) return;
    int hd   = idx & 3;
    int node = idx >> 2;
    const float* hp = hbuf + (size_t)node * OUTF + hd * FH;
    const float* as = attn + hd * (2 * FH + 1);
    float s = 0.f, d = 0.f;
    for (int f = 0; f < FH; ++f) {
        float v = hp[f];
        s += v * as[f];
        d += v * as[FH + f];
    }
    srcb[idx] = s;
    dstb[idx] = d;
}

// ---------------------------------------------------------------------------
// Kernel 3: attention. One block = (b, 16-row i-block), 8 waves.
//  Pass 1: online softmax stats (max,sum) per (row, head).
//  Per 128-wide j-chunk: pass 2 computes alpha (NT global store + LDS tile),
//  then 8 waves accumulate out = alpha @ h with f32 WMMA.
// ---------------------------------------------------------------------------
__global__ __launch_bounds__(256) void k_attn(const float* __restrict__ adj,
                                              const float* __restrict__ hbuf,
                                              const float* __restrict__ srcb,
                                              const float* __restrict__ dstb,
                                              const float* __restrict__ attn,
                                              float* __restrict__ outp,
                                              float* __restrict__ alpha) {
    __shared__ float s_alpha[H][16][APAD];      // ~33 KB alpha chunk tile
    __shared__ float red_m[16][H][16];          // pass-1 partials
    __shared__ float red_s[16][H][16];
    __shared__ float s_m[16][H];
    __shared__ float s_inv[16][H];              // 1/sum (0 if empty row)
    __shared__ float s_src[16][H];

    const int tid = threadIdx.x;
    const int b   = blockIdx.x >> 6;            // N/16 = 64 row-blocks / batch
    const int i0  = (blockIdx.x & 63) * 16;

    if (tid < 64) {                             // stage src for this row block
        int r = tid >> 2, hd = tid & 3;
        s_src[r][hd] = srcb[(size_t)(b * N + i0 + r) * H + hd];
    }
    __syncthreads();

    // ---- Pass 1: row = tid>>4, sub = tid&15, all 4 heads per thread ----
    {
        const int row = tid >> 4, sub = tid & 15;
        float ae[H], sr[H], m[H], ss[H];
        for (int hd = 0; hd < H; ++hd) {
            ae[hd] = attn[hd * (2 * FH + 1) + 2 * FH];
            sr[hd] = s_src[row][hd];
            m[hd]  = -1e30f;
            ss[hd] = 0.f;
        }
        const float* adj_row = adj + ((size_t)(b * N + i0 + row)) * N;
        for (int j = sub; j < N; j += 16) {
            float av = adj_row[j];
            if (av != 0.f) {
                const float* dp = dstb + (size_t)(b * N + j) * H;
                for (int hd = 0; hd < H; ++hd) {
                    float e = sr[hd] + dp[hd] + av * ae[hd];
                    e = (e >= 0.f) ? e : NEG_SLOPE * e;
                    float mn = fmaxf(m[hd], e);
                    ss[hd] = ss[hd] * __expf(m[hd] - mn) + __expf(e - mn);
                    m[hd] = mn;
                }
            }
        }
        for (int hd = 0; hd < H; ++hd) {
            red_m[row][hd][sub] = m[hd];
            red_s[row][hd][sub] = ss[hd];
        }
    }
    __syncthreads();
    if (tid < 64) {                             // combine 16 partials
        int r = tid >> 2, hd = tid & 3;
        float M = -1e30f;
        for (int p = 0; p < 16; ++p) M = fmaxf(M, red_m[r][hd][p]);
        float S = 0.f;
        for (int p = 0; p < 16; ++p) S += red_s[r][hd][p] * __expf(red_m[r][hd][p] - M);
        s_m[r][hd]   = M;
        s_inv[r][hd] = (S > 0.f) ? (1.f / S) : 0.f;
    }
    __syncthreads();

    // ---- Chunked pass 2 + WMMA aggregation ----
    const int wid   = tid >> 5;
    const int L     = tid & 31;
    const int headw = wid >> 1;                 // wave's head
    const int ft    = wid & 1;                  // f half (0/1)
    const int mr    = L & 15;
    const int kb    = (L >> 4) * 2;
    const int hi    = L >> 4;

    const int jj  = tid >> 2;                   // pass-2 mapping (coalesced)
    const int hd2 = tid & 3;
    const float aev = attn[hd2 * (2 * FH + 1) + 2 * FH];

    v8f acc = {};
    for (int c0 = 0; c0 < N; c0 += NCHUNK) {
        // compute alpha chunk: 2 halves x 16 rows
        for (int half = 0; half < 2; ++half) {
            int jl = half * 64 + jj;
            int j  = c0 + jl;
            float dv = dstb[(size_t)(b * N + j) * H + hd2];
            for (int r = 0; r < 16; ++r) {
                float av = adj[((size_t)(b * N + i0 + r)) * N + j];
                float al = 0.f;
                if (av != 0.f) {                // av!=0 implies row sum > 0
                    float e = s_src[r][hd2] + dv + av * aev;
                    e  = (e >= 0.f) ? e : NEG_SLOPE * e;
                    al = __expf(e - s_m[r][hd2]) * s_inv[r][hd2];
                }
                s_alpha[hd2][r][jl] = al;
                // 134 MB write-once stream: keep it out of L2 (NT hint)
                __builtin_nontemporal_store(
                    al, &alpha[(((size_t)(b * N + i0 + r)) * N + j) * H + hd2]);
            }
        }
        __syncthreads();

        // WMMA: out_tile(16 rows x 16 f) += alpha(16 x NCHUNK) @ h(NCHUNK x 16)
        const float* hB = hbuf + ((size_t)(b * N + c0)) * OUTF + headw * FH + ft * 16;
        for (int k0 = 0; k0 < NCHUNK; k0 += 4) {
            v2f a, bb;
            a.x = s_alpha[headw][mr][k0 + kb];
            a.y = s_alpha[headw][mr][k0 + kb + 1];
            int k = k0 + kb;
            bb.x = hB[(size_t)k * OUTF + (L & 15)];
            bb.y = hB[(size_t)(k + 1) * OUTF + (L & 15)];
            acc = __builtin_amdgcn_wmma_f32_16x16x4_f32(false, a, false, bb,
                                                        (short)0, acc, false, false);
        }
        __syncthreads();
    }

    for (int r = 0; r < 8; ++r) {
        int m = r + 8 * hi;
        __builtin_nontemporal_store(
            acc[r],
            &outp[((size_t)(b * N + i0 + m)) * OUTF + headw * FH + ft * 16 + (L & 15)]);
    }
}

// ---------------------------------------------------------------------------
extern "C" void kernel_launch(void* const* d_in, const int* in_sizes, int n_in,
                              void* d_out, int out_size, void* d_ws, size_t ws_size,
                              hipStream_t stream) {
    const float* x    = (const float*)d_in[0];   // (B,N,IN_F)
    const float* adj  = (const float*)d_in[1];   // (B,N,N)
    const float* W    = (const float*)d_in[2];   // (H*FH, IN_F)
    const float* attn = (const float*)d_in[3];   // (H, 2*FH+1)

    float* outp      = (float*)d_out;                       // (B,N,H*FH)
    float* alpha     = outp + (size_t)B * N * OUTF;         // (B,N,N,H)

    float* hbuf = (float*)d_ws;                             // (B*N, 128)
    float* srcb = hbuf + (size_t)B * N * OUTF;              // (B*N, H)
    float* dstb = srcb + (size_t)B * N * H;                 // (B*N, H)

    k_h   <<<(B * N) / 16, 256, 0, stream>>>(x, W, hbuf);
    k_sd  <<<(B * N * H + 255) / 256, 256, 0, stream>>>(hbuf, attn, srcb, dstb);
    k_attn<<<B * (N / 16), 256, 0, stream>>>(adj, hbuf, srcb, dstb, attn, outp, alpha);
}